// Model_38714835206518
// MI455X (gfx1250) — compile-verified
//
#include <hip/hip_runtime.h>
#include <math.h>

// ---------------------------------------------------------------------------
// CDNA5 (gfx1250) wave32 WMMA fp32 GEMM path.
// One wave computes a 64x32 C macro-tile = 4x2 grid of 16x16 WMMA tiles.
// V_WMMA_F32_16X16X4_F32 keeps full fp32 precision (reference is f32; the
// workload is memory-bound: ~110MB / 23.3TB/s >> 57 GFLOP at fp32 matrix rate).
// Per 4-wide K step: 4 x b64 (A) + 4 x b32 (B) loads feed 8 WMMAs
// -> ~1 VMEM instruction per WMMA, 2x (A) / 4x (B) register reuse of each fragment.
// GUARD=false instantiations (all dominant GEMMs: tile-aligned M,N) have no
// clamps/selects/exec-masking at all; GUARD=true handles edge tiles with
// selects only, so EXEC is all-ones at every WMMA.
// ---------------------------------------------------------------------------
typedef __attribute__((ext_vector_type(2))) float v2f;
typedef __attribute__((ext_vector_type(8))) float v8f;

#define CEILDIV(a, b) (((a) + (b) - 1) / (b))

#define MT 4  // 16-row subtiles per wave
#define NT 2  // 16-col subtiles per wave

// C[M,N] = op(A[M,K] @ B[K,N] (+bias))  (row-major; TRANSB => B is [N,K])
template <bool RELU, bool BIAS, bool TRANSB, bool GUARD>
__global__ void __launch_bounds__(32)
wmma_gemm(const float* __restrict__ A, const float* __restrict__ B,
          const float* __restrict__ bias, float* __restrict__ C,
          int M, int N, int K,
          long long sA, long long sB, long long sC) {
  A += (long long)blockIdx.z * sA;
  B += (long long)blockIdx.z * sB;
  C += (long long)blockIdx.z * sC;

  const int lane = threadIdx.x;      // 0..31
  const int half = lane >> 4;        // 0: K+0/1, 1: K+2/3
  const int r    = lane & 15;
  const int row0 = blockIdx.y * (16 * MT);
  const int col0 = blockIdx.x * (16 * NT);

  bool aOk[MT];
  const float* aP[MT];
#pragma unroll
  for (int i = 0; i < MT; ++i) {
    int aRow = row0 + 16 * i + r;
    aOk[i] = GUARD ? (aRow < M) : true;
    aP[i] = A + (size_t)(aOk[i] ? aRow : 0) * K;   // clamped in-bounds
  }
  bool bOk[NT];
  int  colC[NT];
#pragma unroll
  for (int j = 0; j < NT; ++j) {
    int colR = col0 + 16 * j + r;
    bOk[j] = GUARD ? (colR < N) : true;
    colC[j] = bOk[j] ? colR : 0;
  }

  v8f acc[MT][NT] = {};

  for (int kk = 0; kk < K; kk += 4) {
    const int k0 = kk + (half << 1);

    // Speculative prefetch of the B panel ~64 K-steps ahead (global_prefetch_b8).
    {
      int kpf = k0 + 256;
      kpf = kpf < K ? kpf : K - 2;
      const float* pf = TRANSB ? (B + (size_t)colC[0] * K + kpf)
                               : (B + (size_t)kpf * N + colC[0]);
      __builtin_prefetch((const void*)pf, 0, 3);
    }

    // A 16x4 frags: lanes0-15 rows M, K=k0,k0+1; lanes16-31 K=k0+2,k0+3
    v2f a[MT];
#pragma unroll
    for (int i = 0; i < MT; ++i) {
      v2f av = *(const v2f*)(aP[i] + k0);          // 8B-aligned (K even, k0 even)
      a[i].x = aOk[i] ? av.x : 0.0f;
      a[i].y = aOk[i] ? av.y : 0.0f;
    }
    // B 4x16 frags: VGPR0 lanes0-15 K row k0, lanes16-31 K row k0+2; VGPR1 +1
    v2f b[NT];
#pragma unroll
    for (int j = 0; j < NT; ++j) {
      float b0, b1;
      if (TRANSB) {
        b0 = B[(size_t)colC[j] * K + k0];
        b1 = B[(size_t)colC[j] * K + k0 + 1];
      } else {
        b0 = B[(size_t)k0 * N + colC[j]];
        b1 = B[(size_t)(k0 + 1) * N + colC[j]];
      }
      b[j].x = bOk[j] ? b0 : 0.0f;
      b[j].y = bOk[j] ? b1 : 0.0f;
    }
#pragma unroll
    for (int i = 0; i < MT; ++i)
#pragma unroll
      for (int j = 0; j < NT; ++j)
        acc[i][j] = __builtin_amdgcn_wmma_f32_16x16x4_f32(
            false, a[i], false, b[j], (short)0, acc[i][j], false, false);
  }

  float bv[NT];
#pragma unroll
  for (int j = 0; j < NT; ++j) bv[j] = BIAS ? bias[colC[j]] : 0.0f;

#pragma unroll
  for (int i = 0; i < MT; ++i) {
#pragma unroll
    for (int j = 0; j < NT; ++j) {
      int colR = col0 + 16 * j + r;
#pragma unroll
      for (int v = 0; v < 8; ++v) {
        // C/D layout: VGPR v -> row v (lanes0-15) / row v+8 (lanes16-31)
        int row = row0 + 16 * i + v + (half << 3);
        float x = acc[i][j][v];
        if (BIAS) x += bv[j];
        if (RELU) x = fmaxf(x, 0.0f);
        if (!GUARD || (row < M && bOk[j])) C[(size_t)row * N + colR] = x;
      }
    }
  }
}

// ---------------------------------------------------------------------------
// Problem constants
// ---------------------------------------------------------------------------
#define B_    32
#define KOBJ  100
#define NS    16
#define NK    8
#define EMB   1024
#define FEAT  2052
#define HID   1024
#define OUT_  3000
#define COMB  512
#define BK    (B_ * KOBJ)   // 3200

// bb centres from image[:,:,2048:2052]
__global__ void centres_kernel(const float* __restrict__ image, float* __restrict__ ctr) {
  int i = blockIdx.x * blockDim.x + threadIdx.x;
  if (i >= BK) return;
  const float* bb = image + (size_t)i * FEAT + (FEAT - 4);
  float x1 = bb[0], y1 = bb[1], x2 = bb[2], y2 = bb[3];
  ctr[i * 2 + 0] = x1 + 0.5f * (x2 - x1);
  ctr[i * 2 + 1] = y1 + 0.5f * (y2 - y1);
}

// h[row,c] = relu(h_img[row,c] + qpart[row/100, c])   (qpart includes b_e1)
__global__ void add_rowvec_relu(float* __restrict__ h, const float* __restrict__ qpart) {
  int i = blockIdx.x * blockDim.x + threadIdx.x;
  if (i >= BK * COMB) return;
  int row = i / COMB;
  int b = row / KOBJ;
  int c = i - row * COMB;
  h[i] = fmaxf(h[i] + qpart[b * COMB + c], 0.0f);
}

// per (b,k) row: top-16 of adj row (desc, ties -> lower index first) + softmax
__global__ void topk_softmax(const float* __restrict__ adj,
                             int* __restrict__ topi, float* __restrict__ aout) {
  int row = blockIdx.x * blockDim.x + threadIdx.x;
  if (row >= BK) return;
  int b = row / KOBJ;
  const float* p = adj + (size_t)b * KOBJ * KOBJ + (size_t)(row - b * KOBJ) * KOBJ;
  float v[NS];
  int   idx[NS];
#pragma unroll
  for (int s = 0; s < NS; ++s) { v[s] = -INFINITY; idx[s] = 0; }
  for (int j = 0; j < KOBJ; ++j) {
    float x = p[j];
    if (x > v[NS - 1]) {
      int q = NS - 1;
      while (q > 0 && x > v[q - 1]) { v[q] = v[q - 1]; idx[q] = idx[q - 1]; --q; }
      v[q] = x; idx[q] = j;
    }
  }
  float mx = v[0];
  float e[NS], sum = 0.0f;
#pragma unroll
  for (int s = 0; s < NS; ++s) { e[s] = expf(v[s] - mx); sum += e[s]; }
  float inv = 1.0f / sum;
#pragma unroll
  for (int s = 0; s < NS; ++s) {
    topi[row * NS + s] = idx[s];
    aout[row * NS + s] = e[s] * inv;
  }
}

// Gaussian mixture weights, normalized over NK; optionally scaled by softmax a.
// One thread per (b,k,s); writes wn[bk*NS*NK + s*NK + m].
__global__ void gauss_kernel(const float* __restrict__ ctr, const int* __restrict__ topi,
                             const float* __restrict__ a_or_null,
                             const float* __restrict__ mr, const float* __restrict__ mt,
                             const float* __restrict__ pr, const float* __restrict__ pt,
                             float* __restrict__ wn) {
  int i = blockIdx.x * blockDim.x + threadIdx.x;  // 0..BK*NS-1
  if (i >= BK * NS) return;
  int bk = i >> 4;
  int b = bk / KOBJ;
  int j = topi[i];
  float dx = ctr[bk * 2 + 0] - ctr[(b * KOBJ + j) * 2 + 0];
  float dy = ctr[bk * 2 + 1] - ctr[(b * KOBJ + j) * 2 + 1];
  float rho = sqrtf(dx * dx + dy * dy);
  float th  = atan2f(dx, dy);           // reference: arctan2(d0, d1)
  const float TWO_PI = 6.28318530717958647692f;
  float w[NK], sum = 0.0f;
#pragma unroll
  for (int m = 0; m < NK; ++m) {
    float dr = rho - mr[m];
    float wr = expf(-0.5f * dr * dr / (1e-14f + pr[m] * pr[m]));
    float fa = fabsf(th - mt[m]);
    float sa = fabsf(TWO_PI - fa);
    float mn = fminf(fa, sa);
    float wt = expf(-0.5f * mn * mn / (1e-14f + pt[m] * pt[m]));
    float ww = wr * wt;
    ww = (ww != ww) ? 0.0f : ww;        // isnan -> 0
    w[m] = ww; sum += ww;
  }
  float scale = 1.0f / sum;
  if (a_or_null) scale *= a_or_null[i];
#pragma unroll
  for (int m = 0; m < NK; ++m) wn[(size_t)i * NK + m] = w[m] * scale;
}

// out[bk, m*No+o] = relu( sum_s wn[bk,s,m] * P[m, b*100+topi[bk,s], o] )
// blockDim.x == No (256 for conv1, 128 for conv2); grid = BK blocks.
__global__ void gather_combine(const float* __restrict__ P, const float* __restrict__ wn,
                               const int* __restrict__ topi, float* __restrict__ out,
                               int No) {
  __shared__ float sw[NS * NK];
  __shared__ int   si[NS];
  const int bk = blockIdx.x;
  const int b  = bk / KOBJ;
  const int t  = threadIdx.x;
  if (t < NS * NK) sw[t] = wn[(size_t)bk * NS * NK + t];
  if (t < NS) si[t] = topi[bk * NS + t];
  __syncthreads();
  const size_t mStride = (size_t)BK * No;
  const float* Pb = P + (size_t)b * KOBJ * No;
#pragma unroll
  for (int m = 0; m < NK; ++m) {
    float acc = 0.0f;
#pragma unroll
    for (int s = 0; s < NS; ++s)
      acc = fmaf(sw[s * NK + m], Pb[m * mStride + (size_t)si[s] * No + t], acc);
    out[(size_t)bk * (NK * No) + m * No + t] = fmaxf(acc, 0.0f);
  }
}

// hh[b,c] = relu(qenc[b,c]) * max_k g2pre[b,k,c]
__global__ void max_relu_mul(const float* __restrict__ g2pre,
                             const float* __restrict__ qenc, float* __restrict__ hh) {
  int i = blockIdx.x * blockDim.x + threadIdx.x;
  if (i >= B_ * HID) return;
  int b = i >> 10;
  int c = i & (HID - 1);
  const float* p = g2pre + (size_t)b * KOBJ * HID + c;
  float m = p[0];
  for (int k = 1; k < KOBJ; ++k) m = fmaxf(m, p[(size_t)k * HID]);
  hh[i] = fmaxf(qenc[i], 0.0f) * m;
}

// ---------------------------------------------------------------------------
// Workspace layout (float offsets)
// ---------------------------------------------------------------------------
static const size_t OFF_QENC  = 0;                          // 32*1024
static const size_t OFF_QPART = OFF_QENC  + 32768;          // 32*512
static const size_t OFF_H     = OFF_QPART + 16384;          // 3200*512 (reused as g2pre with H2)
static const size_t OFF_H2    = OFF_H     + 1638400;        // 3200*512
static const size_t OFF_CTR   = OFF_H2    + 1638400;        // 3200*2
static const size_t OFF_TOPI  = OFF_CTR   + 6400;           // 3200*16 (int)
static const size_t OFF_A     = OFF_TOPI  + 51200;          // 3200*16
static const size_t OFF_WN1   = OFF_A     + 51200;          // 3200*16*8
static const size_t OFF_WN2   = OFF_WN1   + 409600;         // 3200*16*8
static const size_t OFF_P1    = OFF_WN2   + 409600;         // 8*3200*256 (reused as P2)
static const size_t OFF_G1    = OFF_P1    + 6553600;        // 3200*2048
static const size_t OFF_HH    = OFF_G1    + 6553600;        // 32*1024
static const size_t OFF_H1    = OFF_HH    + 32768;          // 32*3000
// total ~17.5M floats (~70 MB)

extern "C" void kernel_launch(void* const* d_in, const int* in_sizes, int n_in,
                              void* d_out, int out_size, void* d_ws, size_t ws_size,
                              hipStream_t stream) {
  (void)in_sizes; (void)n_in; (void)out_size; (void)ws_size;
  const float* question = (const float*)d_in[0];
  const float* image    = (const float*)d_in[1];
  /* d_in[2] = K (always 100) */
  const float* W_lproj  = (const float*)d_in[3];
  const float* b_lproj  = (const float*)d_in[4];
  const float* W_e1     = (const float*)d_in[5];
  const float* b_e1     = (const float*)d_in[6];
  const float* W_e2     = (const float*)d_in[7];
  const float* b_e2     = (const float*)d_in[8];
  const float* conv_w1  = (const float*)d_in[9];
  const float* mr1      = (const float*)d_in[10];
  const float* mt1      = (const float*)d_in[11];
  const float* pr1      = (const float*)d_in[12];
  const float* pt1      = (const float*)d_in[13];
  const float* conv_w2  = (const float*)d_in[14];
  const float* mr2      = (const float*)d_in[15];
  const float* mt2      = (const float*)d_in[16];
  const float* pr2      = (const float*)d_in[17];
  const float* pt2      = (const float*)d_in[18];
  const float* W_o1     = (const float*)d_in[19];
  const float* b_o1     = (const float*)d_in[20];
  const float* W_o2     = (const float*)d_in[21];
  const float* b_o2     = (const float*)d_in[22];

  float* out    = (float*)d_out;
  float* logits = out;                 // 32*3000
  float* adj    = out + B_ * OUT_;     // 32*100*100

  float* ws    = (float*)d_ws;
  float* qenc  = ws + OFF_QENC;
  float* qpart = ws + OFF_QPART;
  float* h     = ws + OFF_H;
  float* h2    = ws + OFF_H2;
  float* ctr   = ws + OFF_CTR;
  int*   topi  = (int*)(ws + OFF_TOPI);
  float* aw    = ws + OFF_A;
  float* wn1   = ws + OFF_WN1;
  float* wn2   = ws + OFF_WN2;
  float* P1    = ws + OFF_P1;
  float* g1    = ws + OFF_G1;
  float* P2    = ws + OFF_P1;          // reuse (P1 dead after combine1)
  float* g2pre = ws + OFF_H;           // reuse (h/h2 dead after adj)
  float* hh    = ws + OFF_HH;
  float* h1    = ws + OFF_H1;

  const dim3 wave(32);
  const int TM = 16 * MT;  // 64 rows per wave
  const int TN = 16 * NT;  // 32 cols per wave

  // bb centres
  centres_kernel<<<CEILDIV(BK, 256), 256, 0, stream>>>(image, ctr);

  // qenc = question @ W_lproj + b_lproj            (32x1024x1024)  [M edge]
  wmma_gemm<false, true, false, true><<<dim3(CEILDIV(HID, TN), CEILDIV(B_, TM), 1), wave, 0, stream>>>(
      question, W_lproj, b_lproj, qenc, B_, HID, EMB, 0, 0, 0);

  // qpart = qenc @ W_e1[2052:,:] + b_e1            (32x1024x512)   [M edge]
  wmma_gemm<false, true, false, true><<<dim3(CEILDIV(COMB, TN), CEILDIV(B_, TM), 1), wave, 0, stream>>>(
      qenc, W_e1 + (size_t)FEAT * COMB, b_e1, qpart, B_, COMB, HID, 0, 0, 0);

  // h_img = image @ W_e1[:2052,:]                  (3200x2052x512) [aligned]
  wmma_gemm<false, false, false, false><<<dim3(CEILDIV(COMB, TN), CEILDIV(BK, TM), 1), wave, 0, stream>>>(
      image, W_e1, nullptr, h, BK, COMB, FEAT, 0, 0, 0);

  // h = relu(h_img + qpart[b])
  add_rowvec_relu<<<CEILDIV(BK * COMB, 256), 256, 0, stream>>>(h, qpart);

  // h2 = relu(h @ W_e2 + b_e2)                     (3200x512x512)  [aligned]
  wmma_gemm<true, true, false, false><<<dim3(CEILDIV(COMB, TN), CEILDIV(BK, TM), 1), wave, 0, stream>>>(
      h, W_e2, b_e2, h2, BK, COMB, COMB, 0, 0, 0);

  // adj[b] = h2_b @ h2_b^T                         (batched 100x100x512) [edges]
  wmma_gemm<false, false, true, true><<<dim3(CEILDIV(KOBJ, TN), CEILDIV(KOBJ, TM), B_), wave, 0, stream>>>(
      h2, h2, nullptr, adj, KOBJ, KOBJ, COMB,
      (long long)KOBJ * COMB, (long long)KOBJ * COMB, (long long)KOBJ * KOBJ);

  // top-16 + softmax
  topk_softmax<<<CEILDIV(BK, 128), 128, 0, stream>>>(adj, topi, aw);

  // Gaussian kernel weights (conv1 scaled by softmax a; conv2 unscaled)
  gauss_kernel<<<CEILDIV(BK * NS, 256), 256, 0, stream>>>(
      ctr, topi, aw, mr1, mt1, pr1, pt1, wn1);
  gauss_kernel<<<CEILDIV(BK * NS, 256), 256, 0, stream>>>(
      ctr, topi, nullptr, mr2, mt2, pr2, pt2, wn2);

  // P1[m] = image @ conv_w1[m]                     (8 x 3200x2052x256) [aligned]
  wmma_gemm<false, false, false, false><<<dim3(CEILDIV(256, TN), CEILDIV(BK, TM), NK), wave, 0, stream>>>(
      image, conv_w1, nullptr, P1, BK, 256, FEAT,
      0, (long long)FEAT * 256, (long long)BK * 256);

  // g1 = relu(gathered weighted sum of P1)         -> (3200 x 2048)
  gather_combine<<<BK, 256, 0, stream>>>(P1, wn1, topi, g1, 256);

  // P2[m] = g1 @ conv_w2[m]                        (8 x 3200x2048x128) [aligned]
  wmma_gemm<false, false, false, false><<<dim3(CEILDIV(128, TN), CEILDIV(BK, TM), NK), wave, 0, stream>>>(
      g1, conv_w2, nullptr, P2, BK, 128, 2 * HID,
      0, (long long)(2 * HID) * 128, (long long)BK * 128);

  // g2pre = relu(gathered sum of P2)               -> (3200 x 1024)
  gather_combine<<<BK, 128, 0, stream>>>(P2, wn2, topi, g2pre, 128);

  // hh = relu(qenc) * max_k g2pre
  max_relu_mul<<<CEILDIV(B_ * HID, 256), 256, 0, stream>>>(g2pre, qenc, hh);

  // h1 = relu(hh @ W_o1 + b_o1)                    (32x1024x3000)  [edges]
  wmma_gemm<true, true, false, true><<<dim3(CEILDIV(OUT_, TN), CEILDIV(B_, TM), 1), wave, 0, stream>>>(
      hh, W_o1, b_o1, h1, B_, OUT_, HID, 0, 0, 0);

  // logits = h1 @ W_o2 + b_o2                      (32x3000x3000)  [edges]
  wmma_gemm<false, true, false, true><<<dim3(CEILDIV(OUT_, TN), CEILDIV(B_, TM), 1), wave, 0, stream>>>(
      h1, W_o2, b_o2, logits, B_, OUT_, OUT_, 0, 0, 0);
}